// IdentifyLay_31181462569492
// MI455X (gfx1250) — compile-verified
//
#include <hip/hip_runtime.h>

typedef float v2f __attribute__((ext_vector_type(2)));
typedef float v8f __attribute__((ext_vector_type(8)));

static constexpr int NPTS = 1024;
static constexpr int DIM  = 64;
static constexpr int NN   = NPTS * NPTS;
#define EPSV 1e-12f

// ---------------------------------------------------------------------------
// Kernel 1: per-column L1 sums  s[d] = sum_{i,j} |z[i,d] - z[j,d]|
// grid: 128 blocks (2 z's x 64 columns), 256 threads
// ---------------------------------------------------------------------------
__global__ __launch_bounds__(256)
void colsum_kernel(const float* __restrict__ z1, const float* __restrict__ z2,
                   float* __restrict__ wsS) {
  const int b  = blockIdx.x;
  const int zi = b >> 6;
  const int d  = b & 63;
  const float* __restrict__ z = zi ? z2 : z1;

  __shared__ float col[NPTS];
  __shared__ float red[256];
  const int t = threadIdx.x;

  #pragma unroll
  for (int u = 0; u < 4; ++u) col[t + u * 256] = z[(t + u * 256) * DIM + d];
  __syncthreads();

  float a0 = 0.f, a1 = 0.f, a2 = 0.f, a3 = 0.f;
  const float r0 = col[t], r1 = col[t + 256], r2 = col[t + 512], r3 = col[t + 768];
  for (int j = 0; j < NPTS; ++j) {
    const float c = col[j];
    a0 += fabsf(r0 - c); a1 += fabsf(r1 - c);
    a2 += fabsf(r2 - c); a3 += fabsf(r3 - c);
  }
  red[t] = (a0 + a1) + (a2 + a3);
  __syncthreads();
  for (int off = 128; off > 0; off >>= 1) {
    if (t < off) red[t] += red[t + off];
    __syncthreads();
  }
  if (t == 0) wsS[zi * 64 + d] = red[0];
}

// ---------------------------------------------------------------------------
// Kernel 2: A = z @ (diag(1/s) * W1)   via V_WMMA_F32_16X16X4_F32
// grid: 128 blocks (2 z's x 64 row-tiles of 16), 32 threads (1 wave)
// ---------------------------------------------------------------------------
__global__ __launch_bounds__(32)
void gemm_kernel(const float* __restrict__ z1, const float* __restrict__ z2,
                 const float* __restrict__ W1, const float* __restrict__ wsS,
                 float* __restrict__ wsA) {
  const int b  = blockIdx.x;
  const int zi = b >> 6;
  const int rt = b & 63;
  const float* __restrict__ z = zi ? z2 : z1;

  __shared__ float sInv[64];
  const int L = threadIdx.x;
  sInv[L]      = 1.0f / fmaxf(wsS[zi * 64 + L],      EPSV);
  sInv[L + 32] = 1.0f / fmaxf(wsS[zi * 64 + L + 32], EPSV);
  __syncthreads();

  const int hlf  = L >> 4;      // lane half: K offset 0 or 2
  const int m    = L & 15;      // row within tile / column within n-tile
  const int row0 = rt * 16;

  // A fragments: 16x4 f32 layout — lanes 0-15 hold K=4kk+{0,1}, lanes 16-31 K=4kk+{2,3}
  v2f afrag[16];
  #pragma unroll
  for (int kk = 0; kk < 16; ++kk) {
    const float* p = z + (row0 + m) * DIM + kk * 4 + 2 * hlf;
    afrag[kk].x = p[0];
    afrag[kk].y = p[1];
  }

  #pragma unroll
  for (int nt = 0; nt < 4; ++nt) {
    const int n0 = nt * 16;
    v8f c = {0.f, 0.f, 0.f, 0.f, 0.f, 0.f, 0.f, 0.f};
    #pragma unroll
    for (int kk = 0; kk < 16; ++kk) {
      const int k0 = kk * 4 + 2 * hlf;
      v2f bfrag;
      bfrag.x = W1[(k0 + 0) * DIM + n0 + m] * sInv[k0 + 0];
      bfrag.y = W1[(k0 + 1) * DIM + n0 + m] * sInv[k0 + 1];
      c = __builtin_amdgcn_wmma_f32_16x16x4_f32(
              /*neg_a=*/false, afrag[kk], /*neg_b=*/false, bfrag,
              /*c_mod=*/(short)0, c, /*reuse_a=*/false, /*reuse_b=*/false);
    }
    // C/D 16x16 f32 layout: VGPR v -> row v (lanes 0-15) or v+8 (lanes 16-31)
    float* Ao = wsA + zi * (NPTS * DIM) + n0 + m;
    #pragma unroll
    for (int v = 0; v < 8; ++v) Ao[(row0 + hlf * 8 + v) * DIM] = c[v];
  }
}

// ---------------------------------------------------------------------------
// Kernel 3: o[i,j] = relu(b2 + sum_k relu(A[i,k]-A[j,k]+b1[k]) * w2[k])
// writes unnormalized o into out planes 0/1, row sums into wsRow.
// grid: 128 blocks (2 z's x 64 i-tiles of 16 rows), 512 threads (16 waves)
// ---------------------------------------------------------------------------
__global__ __launch_bounds__(512)
void pair_kernel(const float* __restrict__ wsA, const float* __restrict__ b1,
                 const float* __restrict__ W2, const float* __restrict__ b2,
                 float* __restrict__ out, float* __restrict__ wsRow) {
  const int b  = blockIdx.x;
  const int zi = b >> 6;
  const int i0 = (b & 63) * 16;
  const float* __restrict__ A = wsA + zi * (NPTS * DIM);

  __shared__ float sAi[16 * 64];       // A rows i0..i0+15, with b1 folded in
  __shared__ float sAj[32 * 65];       // padded j-tile (bank-conflict free)
  __shared__ float sW2[64];

  const int t = threadIdx.x;
  sAi[t]       = A[i0 * DIM + t]       + b1[t & 63];
  sAi[t + 512] = A[i0 * DIM + t + 512] + b1[(t + 512) & 63];
  if (t < 64) sW2[t] = W2[t];
  const float b2v = b2[0];

  const int w = t >> 5, lane = t & 31;
  const int i = i0 + w;
  float rowAcc = 0.0f;

  for (int jc = 0; jc < 32; ++jc) {
    __syncthreads();
    #pragma unroll
    for (int u = 0; u < 4; ++u) {
      const int idx = t + u * 512;
      const int jr = idx >> 6, k = idx & 63;
      sAj[jr * 65 + k] = A[(jc * 32 + jr) * DIM + k];
    }
    __syncthreads();

    float acc = b2v;
    const float* ai = &sAi[w * 64];
    const float* aj = &sAj[lane * 65];
    #pragma unroll
    for (int k = 0; k < 64; ++k) {
      float x = ai[k] - aj[k];
      x = fmaxf(x, 0.0f);
      acc = fmaf(x, sW2[k], acc);
    }
    const float o = fmaxf(acc, 0.0f);
    out[zi * NN + i * NPTS + jc * 32 + lane] = o;
    rowAcc += o;
  }

  // deterministic wave32 reduction for the row sum
  #pragma unroll
  for (int off = 16; off > 0; off >>= 1)
    rowAcc += __shfl_xor(rowAcc, off, 32);
  if (lane == 0) wsRow[zi * NPTS + i] = rowAcc;
}

// ---------------------------------------------------------------------------
// Kernel 4: row-wise L1 normalize planes 0/1 in place; duplicate to planes 2/3
// ---------------------------------------------------------------------------
__global__ __launch_bounds__(256)
void norm_kernel(float* __restrict__ out, const float* __restrict__ wsRow) {
  const int gid = blockIdx.x * 256 + threadIdx.x;   // 0 .. 2*NN-1
  const int p   = gid >> 20;                        // NN == 1<<20
  const int rem = gid & (NN - 1);
  const int i   = rem >> 10;
  const float inv = 1.0f / fmaxf(wsRow[p * NPTS + i], EPSV);
  const float v = out[gid] * inv;
  out[gid] = v;
  out[(p + 2) * NN + rem] = v;
}

// ---------------------------------------------------------------------------
extern "C" void kernel_launch(void* const* d_in, const int* in_sizes, int n_in,
                              void* d_out, int out_size, void* d_ws, size_t ws_size,
                              hipStream_t stream) {
  const float* z1 = (const float*)d_in[0];
  const float* z2 = (const float*)d_in[1];
  const float* W1 = (const float*)d_in[2];
  const float* b1 = (const float*)d_in[3];
  const float* W2 = (const float*)d_in[4];
  const float* b2 = (const float*)d_in[5];
  float* out = (float*)d_out;

  float* wsS   = (float*)d_ws;           // 128 floats (padded to 256)
  float* wsA   = wsS + 256;              // 2 * 1024 * 64 floats
  float* wsRow = wsA + 2 * NPTS * DIM;   // 2048 floats

  colsum_kernel<<<128, 256, 0, stream>>>(z1, z2, wsS);
  gemm_kernel<<<128, 32, 0, stream>>>(z1, z2, W1, wsS, wsA);
  pair_kernel<<<128, 512, 0, stream>>>(wsA, b1, W2, b2, out, wsRow);
  norm_kernel<<<(2 * NN) / 256, 256, 0, stream>>>(out, wsRow);
}